// SSIM_40080634807168
// MI455X (gfx1250) — compile-verified
//
#include <hip/hip_runtime.h>

typedef __attribute__((ext_vector_type(2))) float v2f;
typedef __attribute__((ext_vector_type(8))) float v8f;

#define BATCH   16
#define INW     512
#define CROP    4
#define OUTW    504        // 512 - 2*CROP
#define TILE_X  64
#define TILE_Y  32
#define HALO    5
#define IN_ROWS 42         // TILE_Y + 2*HALO
#define IN_COLS 74         // TILE_X + 2*HALO
#define IN_PITCH 75
#define H_ROWS  44         // 42 used + 2 zero pad rows for K=26,27 of the band
#define H_PITCH 65         // pad to avoid 64-bank conflicts between lane halves
#define GRID_X  8          // ceil(504/64)
#define GRID_Y  16         // ceil(504/32)
#define NBLK    (GRID_X * GRID_Y * BATCH)

__global__ __launch_bounds__(256) void ssim_tile_kernel(const float* __restrict__ img1,
                                                        const float* __restrict__ img2,
                                                        float* __restrict__ partial) {
  __shared__ float sIn[2 * IN_ROWS * IN_PITCH];   // x1, x2 halo tiles
  __shared__ float sH[5 * H_ROWS * H_PITCH];      // horizontal box sums of 5 quantities
  __shared__ float sRed[256];

  const int tid   = threadIdx.x;
  const int tileX = blockIdx.x * TILE_X;
  const int tileY = blockIdx.y * TILE_Y;
  const long planeBase = (long)blockIdx.z * 3 * INW * INW;  // channel 0 of batch b

  // ---- Stage 0: cooperative zero-padded halo load (L2-resident after first touch) ----
  for (int i = tid; i < IN_ROWS * IN_COLS; i += 256) {
    int r = i / IN_COLS, c = i - r * IN_COLS;
    int gy = tileY - HALO + r;
    int gx = tileX - HALO + c;
    float a = 0.0f, b = 0.0f;
    if (gy >= 0 && gy < OUTW && gx >= 0 && gx < OUTW) {
      long idx = planeBase + (long)(gy + CROP) * INW + (gx + CROP);
      a = img1[idx];
      b = img2[idx];
    }
    sIn[0 * IN_ROWS * IN_PITCH + r * IN_PITCH + c] = a;
    sIn[1 * IN_ROWS * IN_PITCH + r * IN_PITCH + c] = b;
  }
  __syncthreads();

  // ---- Stage 1: horizontal 11-tap sums of {x1, x2, x1^2, x2^2, x1*x2} ----
  for (int i = tid; i < H_ROWS * TILE_X; i += 256) {
    int r = i >> 6, c = i & 63;
    float s1 = 0.f, s2 = 0.f, s11 = 0.f, s22 = 0.f, s12 = 0.f;
    if (r < IN_ROWS) {
      const float* p1 = &sIn[0 * IN_ROWS * IN_PITCH + r * IN_PITCH + c];
      const float* p2 = &sIn[1 * IN_ROWS * IN_PITCH + r * IN_PITCH + c];
#pragma unroll
      for (int dx = 0; dx <= 10; ++dx) {
        float a = p1[dx], b = p2[dx];
        s1 += a; s2 += b; s11 += a * a; s22 += b * b; s12 += a * b;
      }
    }
    sH[0 * H_ROWS * H_PITCH + r * H_PITCH + c] = s1;
    sH[1 * H_ROWS * H_PITCH + r * H_PITCH + c] = s2;
    sH[2 * H_ROWS * H_PITCH + r * H_PITCH + c] = s11;
    sH[3 * H_ROWS * H_PITCH + r * H_PITCH + c] = s22;
    sH[4 * H_ROWS * H_PITCH + r * H_PITCH + c] = s12;
  }
  __syncthreads();

  // ---- Stage 2: vertical 11-tap pass via f32 WMMA: D = Band(16x28) * H(28x16) ----
  // A layout (16x4 f32): lanes 0-15 -> M=0..15; VGPR0 holds K=0 (lo lanes)/K=2 (hi),
  // VGPR1 holds K=1/K=3. B (4x16) mirrors with N across lanes, K split in halves.
  const int wave = tid >> 5;
  const int lane = tid & 31;
  const int sx = (wave & 3) * 16;
  const int sy = (wave >> 2) * 16;
  const int mn = lane & 15;           // M for A/D, N for B/D
  const int khi = (lane >= 16) ? 2 : 0;

  v8f zero = {0.f, 0.f, 0.f, 0.f, 0.f, 0.f, 0.f, 0.f};
  v8f acc[5] = {zero, zero, zero, zero, zero};

#pragma unroll
  for (int kb = 0; kb < 7; ++kb) {
    v2f a;
#pragma unroll
    for (int j = 0; j < 2; ++j) {
      int kk = 4 * kb + j + khi;
      // Band matrix: out row m sums H rows sy+m .. sy+m+10
      a[j] = (kk >= mn && kk <= mn + 10) ? 1.0f : 0.0f;
    }
#pragma unroll
    for (int q = 0; q < 5; ++q) {
      v2f bv;
#pragma unroll
      for (int j = 0; j < 2; ++j) {
        int kk = 4 * kb + j + khi;
        bv[j] = sH[q * H_ROWS * H_PITCH + (sy + kk) * H_PITCH + (sx + mn)];
      }
      acc[q] = __builtin_amdgcn_wmma_f32_16x16x4_f32(false, a, false, bv,
                                                     (short)0, acc[q], false, false);
    }
  }

  // ---- Stage 3: pointwise SSIM + masked accumulate ----
  const float inv121 = 1.0f / 121.0f;
  const float C1 = 6.5025f;    // (0.01*255)^2
  const float C2 = 58.5225f;   // (0.03*255)^2
  float lsum = 0.0f;
#pragma unroll
  for (int j = 0; j < 8; ++j) {
    int m = j + ((lane >= 16) ? 8 : 0);          // C/D layout: VGPR j -> M=j (lo)/j+8 (hi)
    int gy = tileY + sy + m;
    int gx = tileX + sx + mn;
    float mu1 = acc[0][j] * inv121;
    float mu2 = acc[1][j] * inv121;
    float mu1s = mu1 * mu1, mu2s = mu2 * mu2, mu12 = mu1 * mu2;
    float sg1 = acc[2][j] * inv121 - mu1s;
    float sg2 = acc[3][j] * inv121 - mu2s;
    float sg12 = acc[4][j] * inv121 - mu12;
    float num = (2.0f * mu12 + C1) * (2.0f * sg12 + C2);
    float den = (mu1s + mu2s + C1) * (sg1 + sg2 + C2);
    float v = num / den;
    lsum += (gy < OUTW && gx < OUTW) ? v : 0.0f;
  }

  // ---- Block reduction -> per-block partial sum ----
  sRed[tid] = lsum;
  __syncthreads();
#pragma unroll
  for (int s = 128; s > 0; s >>= 1) {
    if (tid < s) sRed[tid] += sRed[tid + s];
    __syncthreads();
  }
  if (tid == 0) {
    partial[(blockIdx.z * gridDim.y + blockIdx.y) * gridDim.x + blockIdx.x] = sRed[0];
  }
}

__global__ __launch_bounds__(256) void ssim_reduce_kernel(const float* __restrict__ partial,
                                                          float* __restrict__ out, int n) {
  __shared__ double sRed[256];
  double s = 0.0;
  for (int i = threadIdx.x; i < n; i += 256) s += (double)partial[i];
  sRed[threadIdx.x] = s;
  __syncthreads();
#pragma unroll
  for (int k = 128; k > 0; k >>= 1) {
    if (threadIdx.x < k) sRed[threadIdx.x] += sRed[threadIdx.x + k];
    __syncthreads();
  }
  if (threadIdx.x == 0)
    out[0] = (float)(sRed[0] / ((double)BATCH * (double)OUTW * (double)OUTW));
}

extern "C" void kernel_launch(void* const* d_in, const int* in_sizes, int n_in,
                              void* d_out, int out_size, void* d_ws, size_t ws_size,
                              hipStream_t stream) {
  const float* img1 = (const float*)d_in[0];
  const float* img2 = (const float*)d_in[1];
  float* partial = (float*)d_ws;   // NBLK floats = 8 KB of scratch

  dim3 grid(GRID_X, GRID_Y, BATCH);
  ssim_tile_kernel<<<grid, 256, 0, stream>>>(img1, img2, partial);
  ssim_reduce_kernel<<<1, 256, 0, stream>>>(partial, (float*)d_out, NBLK);
}